// GATLayer_68822555951381
// MI455X (gfx1250) — compile-verified
//
#include <hip/hip_runtime.h>
#include <hip/hip_bf16.h>
#include <math.h>

typedef __attribute__((ext_vector_type(2))) float v2f;
typedef __attribute__((ext_vector_type(8))) float v8f;

#define BSZ 8
#define NN  1024
#define IND 64
#define HD  64
#define LBL 4
#define NEGF (-9.0e15f)
#define SSTRIDE 1028   // rows 16B-aligned; A-frag banks = 4*m (conflict-free)

// ---------------------------------------------------------------------------
// Kernel 1: h = tanh(LN(concat(x,feature) @ W)); also s_src/s_dst projections.
// One 64-thread block per (b,i) row. Tiny GEMM (69 MFLOP) — VALU is fine here.
// ---------------------------------------------------------------------------
__global__ __launch_bounds__(64) void gat_h_kernel(
    const float* __restrict__ x, const float* __restrict__ feat,
    const float* __restrict__ W, const float* __restrict__ a,
    const float* __restrict__ gamma, const float* __restrict__ beta,
    float* __restrict__ hbuf, float* __restrict__ ssrc, float* __restrict__ sdst)
{
    const int row = blockIdx.x;          // b*NN + i
    const int tid = threadIdx.x;         // 0..63 == output channel n
    __shared__ float rowv[IND + 2];
    __shared__ float hrow[HD];

    rowv[tid] = x[row * IND + tid];
    if (tid < 2) rowv[IND + tid] = feat[row * 2 + tid];
    __syncthreads();

    float acc = 0.0f;
    #pragma unroll 6
    for (int k = 0; k < IND + 2; ++k)
        acc = fmaf(rowv[k], W[k * HD + tid], acc);

    hrow[tid] = acc;
    __syncthreads();

    float mu = 0.0f;
    for (int k = 0; k < HD; ++k) mu += hrow[k];
    mu *= (1.0f / HD);
    float var = 0.0f;
    for (int k = 0; k < HD; ++k) { float d = hrow[k] - mu; var = fmaf(d, d, var); }
    var *= (1.0f / HD);

    float hv = tanhf((acc - mu) * rsqrtf(var + 1e-5f) * gamma[tid] + beta[tid]);
    __syncthreads();                    // everyone done reading old hrow
    hrow[tid] = hv;
    hbuf[row * HD + tid] = hv;
    __syncthreads();

    if (tid < 2 * (LBL + 1)) {          // 10 dot products: (label, src/dst half)
        const int l = tid >> 1, half = tid & 1;
        const float* av = a + l * (2 * HD) + half * HD;   // a[l, half*H : half*H+H, 0]
        float s = 0.0f;
        for (int n = 0; n < HD; ++n) s = fmaf(hrow[n], av[n], s);
        const int b = row >> 10, i = row & (NN - 1);
        float* dst = half ? sdst : ssrc;
        dst[(b * (LBL + 1) + l) * NN + i] = s;
    }
}

// ---------------------------------------------------------------------------
// Kernel 2: per (b, 16-row tile): build selected scores (single int4 pass over
// adjs), two-pass softmax with wave32 shuffle reductions (normalization
// deferred to epilogue as a reciprocal multiply), then att @ h via
// V_WMMA_F32_16X16X4_F32 (fp32 accumulate), one 16-wide N-tile per wave.
// ---------------------------------------------------------------------------
__global__ __launch_bounds__(128) void gat_att_kernel(
    const int* __restrict__ adjs, const float* __restrict__ hbuf,
    const float* __restrict__ ssrc, const float* __restrict__ sdst,
    float* __restrict__ out)
{
    __shared__ __align__(16) float S[16][SSTRIDE];  // scores -> exp(scores)
    __shared__ float sdst_sh[(LBL + 1) * NN];       // 20 KB: s_dst[b, :, :]
    __shared__ float ssrc_sh[(LBL + 1) * 16];
    __shared__ float red[64];                       // 16 rows x 4 waves
    __shared__ float rowmax[16];
    __shared__ float rinv[16];

    const int tid  = threadIdx.x;        // 0..127
    const int wave = tid >> 5;
    const int lane = tid & 31;
    const int b    = blockIdx.x >> 6;    // NN/16 = 64 tiles per batch
    const int i0   = (blockIdx.x & 63) << 4;

    // ---- Stage s_dst slab + s_src tile into LDS (coalesced) ----
    const float* sdst_b = sdst + (long)b * (LBL + 1) * NN;
    for (int k = tid; k < (LBL + 1) * NN; k += 128) sdst_sh[k] = sdst_b[k];
    if (tid < (LBL + 1) * 16) {
        const int l = tid >> 4, m = tid & 15;
        ssrc_sh[tid] = ssrc[(b * (LBL + 1) + l) * NN + i0 + m];
    }
    __syncthreads();

    // ---- Phase 1: selection + tanh, int4 pass over adjs; fused row max ----
    const long adjBase = (long)b * LBL * NN * NN;
    const long NN2 = (long)NN * NN;

    auto scoreOf = [&](int a0, int a1, int a2, int a3, int j, int m) -> float {
        int sel = -1;                    // later labels override earlier
        if (a0 > 0) sel = 0;
        if (a1 > 0) sel = 1;
        if (a2 > 0) sel = 2;
        if (a3 > 0) sel = 3;
        if (sel >= 0)
            return tanhf(ssrc_sh[sel * 16 + m] + sdst_sh[sel * NN + j]);
        if (j == i0 + m)
            return tanhf(ssrc_sh[LBL * 16 + m] + sdst_sh[LBL * NN + j]);
        return NEGF;
    };

    for (int m = 0; m < 16; ++m) {
        float lmax = -3.0e38f;
        for (int g = tid; g < NN / 4; g += 128) {      // 2 quads / thread / row
            const int j0 = g << 2;
            const long p = adjBase + (long)(i0 + m) * NN + j0;
            const int4 q0 = *(const int4*)(adjs + p);
            const int4 q1 = *(const int4*)(adjs + p + NN2);
            const int4 q2 = *(const int4*)(adjs + p + 2 * NN2);
            const int4 q3 = *(const int4*)(adjs + p + 3 * NN2);
            float4 sv;
            sv.x = scoreOf(q0.x, q1.x, q2.x, q3.x, j0 + 0, m);
            sv.y = scoreOf(q0.y, q1.y, q2.y, q3.y, j0 + 1, m);
            sv.z = scoreOf(q0.z, q1.z, q2.z, q3.z, j0 + 2, m);
            sv.w = scoreOf(q0.w, q1.w, q2.w, q3.w, j0 + 3, m);
            *(float4*)&S[m][j0] = sv;
            lmax = fmaxf(fmaxf(fmaxf(sv.x, sv.y), fmaxf(sv.z, sv.w)), lmax);
        }
        #pragma unroll
        for (int off = 16; off > 0; off >>= 1)
            lmax = fmaxf(lmax, __shfl_xor(lmax, off, 32));
        if (lane == 0) red[m * 4 + wave] = lmax;
    }
    __syncthreads();
    if (tid < 16)
        rowmax[tid] = fmaxf(fmaxf(red[tid * 4], red[tid * 4 + 1]),
                            fmaxf(red[tid * 4 + 2], red[tid * 4 + 3]));
    __syncthreads();

    // ---- Phase 2: exp in place + row sums (shuffle-reduced) ----
    #pragma unroll
    for (int m = 0; m < 16; ++m) {
        const float rm = rowmax[m];
        float ps = 0.0f;
        #pragma unroll
        for (int k = 0; k < 8; ++k) {
            const int j = tid + (k << 7);
            const float v = __expf(S[m][j] - rm);     // NEG entries -> 0
            S[m][j] = v;
            ps += v;
        }
        #pragma unroll
        for (int off = 16; off > 0; off >>= 1)
            ps += __shfl_xor(ps, off, 32);
        if (lane == 0) red[m * 4 + wave] = ps;
    }
    __syncthreads();
    if (tid < 16)
        rinv[tid] = 1.0f / (red[tid * 4] + red[tid * 4 + 1] +
                            red[tid * 4 + 2] + red[tid * 4 + 3]);
    __syncthreads();

    // ---- Phase 3: D(16x64) = exp(S)(16x1024) @ h_b(1024x64), fp32 WMMA ----
    const int c    = lane & 15;          // N within tile (A: row M; B: col N)
    const int koff = (lane >> 4) << 1;   // 0 | 2 per ISA 16x4 / 4x16 layouts
    const int n0   = wave << 4;
    const float* hp = hbuf + (long)b * NN * HD;

    v8f acc = {};
    for (int kb = 0; kb < NN / 4; ++kb) {
        const int k0 = 4 * kb + koff;
        // A frag: lane<16 holds S[m][k0..k0+1]; lane>=16 holds the K+2 pair.
        const v2f Af = *(const v2f*)&S[c][k0];
        // B frag: h[j=k0, n0+c], h[j=k0+1, n0+c]
        v2f Bf;
        Bf.x = hp[k0 * HD + n0 + c];
        Bf.y = hp[(k0 + 1) * HD + n0 + c];
        acc = __builtin_amdgcn_wmma_f32_16x16x4_f32(
            /*neg_a=*/false, Af, /*neg_b=*/false, Bf,
            /*c_mod=*/(short)0, acc, /*reuse_a=*/false, /*reuse_b=*/false);
    }

    // Epilogue: softmax normalization as reciprocal multiply
    const int mbase = (lane >> 4) << 3;  // C/D: lanes 0-15 -> M=v, 16-31 -> M=v+8
    #pragma unroll
    for (int v = 0; v < 8; ++v) {
        const int m = v + mbase;
        out[((long)b * NN + i0 + m) * HD + n0 + c] = acc[v] * rinv[m];
    }
}

// ---------------------------------------------------------------------------
extern "C" void kernel_launch(void* const* d_in, const int* in_sizes, int n_in,
                              void* d_out, int out_size, void* d_ws, size_t ws_size,
                              hipStream_t stream) {
    const float* x     = (const float*)d_in[0];
    const float* feat  = (const float*)d_in[1];
    const int*   adjs  = (const int*)d_in[2];
    const float* W     = (const float*)d_in[3];
    const float* a     = (const float*)d_in[4];
    const float* gamma = (const float*)d_in[5];
    const float* beta  = (const float*)d_in[6];
    float* out = (float*)d_out;

    float* hbuf = (float*)d_ws;                       // 8*1024*64 f32 = 2 MB
    float* ssrc = hbuf + BSZ * NN * HD;               // 8*5*1024
    float* sdst = ssrc + BSZ * (LBL + 1) * NN;        // 8*5*1024

    gat_h_kernel<<<BSZ * NN, 64, 0, stream>>>(x, feat, W, a, gamma, beta,
                                              hbuf, ssrc, sdst);
    gat_att_kernel<<<BSZ * (NN / 16), 128, 0, stream>>>(adjs, hbuf, ssrc, sdst, out);
}